// NetEval_2207613190845
// MI455X (gfx1250) — compile-verified
//
#include <hip/hip_runtime.h>
#include <cstdint>

// ---------------------------------------------------------------------------
// Piecewise mixture-of-16-experts MLP (2 -> 20 -> 20 -> 20 -> 20 -> 1, SiLU)
// evaluated with fp32 WMMA (V_WMMA_F32_16X16X4_F32) on gfx1250 wave32.
//
// One wave handles one tile of 16 samples (M = 16). HID 20 padded to 32 cols
// (two 16-column accumulator sets). Activations chain D -> A via a per-wave
// LDS tile. Weights are pre-padded/interleaved into d_ws so each lane's
// B operand is one 8-byte load:
//   B VGPR pair layout (16x16x4 f32): lanes 0-15 hold K={0,1}, lanes 16-31
//   hold K={2,3}; element (k, col) stored at ((k/2)*NCOL + col)*2 + (k&1).
//
// Bias folding: activation col 20 is forced to 1.0 and each layer's bias
// lives in weight row k=20 (layer 1: A k=2 slot = 1.0, W1 row k=2 = b1), so
// WMMA produces W^T h + b directly. K loop covers only k=0..23 (6 k-steps).
// SiLU uses the gfx1250 hardware tanh: one v_tanh_f32 + mul + fma.
//
// Round-5 change:
//  * LDS ordering fences are now pure compiler barriers (no s_wait_dscnt).
//    DS instructions from one wave complete in order (ISA 7.3), so the
//    store->load / load->store ordering is guaranteed by hardware; only
//    compiler reordering across the cross-lane LDS handoff must be blocked.
//    The compiler still inserts minimal DScnt waits where loaded data is
//    consumed. Removes ~10 full DS-queue drains per expert iteration.
// ---------------------------------------------------------------------------

typedef float v2f __attribute__((ext_vector_type(2)));
typedef float v8f __attribute__((ext_vector_type(8)));

#define EXP_STRIDE 4096   // floats per expert in the padded weight block
#define W1_OFF 0          // K=4  x 32 cols  -> 128
#define W2_OFF 128        // K=32 x 32 cols  -> 1024
#define W3_OFF 1152
#define W4_OFF 2176
#define W5_OFF 3200       // K=32 x 16 cols  -> 512

#define WPB 8             // waves per block (256 threads)
#define NKS 6             // k-steps for hidden/output layers (K = 24 >= 21)

__device__ __forceinline__ int classify(float v) {
  // bounds: [-4, -0.674, 0, 0.674, 4]; first cell inclusive lower bound,
  // upper bounds inclusive; outside [-4,4] -> -1 (output 0).
  if (v < -4.0f || v > 4.0f) return -1;
  if (v <= -0.674f) return 0;
  if (v <= 0.0f)    return 1;
  if (v <= 0.674f)  return 2;
  return 3;
}

__device__ __forceinline__ float silu(float v) {
#if __has_builtin(__builtin_amdgcn_tanhf)
  // silu(v) = 0.5v + 0.5v*tanh(0.5v): v_mul + v_tanh_f32 + v_fma
  float h = 0.5f * v;
  float t = __builtin_amdgcn_tanhf(h);
  return __builtin_fmaf(h, t, h);
#else
  // v * sigmoid(v): v_mul(-log2e) + v_exp + v_add + v_rcp + v_mul
  float e = __expf(-v);
  return v * __builtin_amdgcn_rcpf(1.0f + e);
#endif
}

__device__ __forceinline__ v8f wmma_f32(v2f a, v2f b, v8f c) {
  // 8 args: (neg_a, A, neg_b, B, c_mod, C, reuse_a, reuse_b)
  return __builtin_amdgcn_wmma_f32_16x16x4_f32(false, a, false, b, (short)0, c,
                                               false, false);
}

// Compiler-only ordering barrier for the cross-lane LDS handoff. Hardware
// keeps same-wave DS instructions in order (DScnt, in-order pipe); we only
// need to stop the compiler from reordering/caching across this point.
__device__ __forceinline__ void lds_order() {
  asm volatile("" ::: "memory");
}

// ---------------------------------------------------------------------------
// Phase 0: pad + interleave weights (biases folded into row k=20, k=2 for L1)
// into d_ws. Runs every launch; d_ws is fully rewritten.
// ---------------------------------------------------------------------------
__global__ void prep_weights(const float* __restrict__ W1, const float* __restrict__ b1,
                             const float* __restrict__ W2, const float* __restrict__ b2,
                             const float* __restrict__ W3, const float* __restrict__ b3,
                             const float* __restrict__ W4, const float* __restrict__ b4,
                             const float* __restrict__ W5, const float* __restrict__ b5,
                             float* __restrict__ wp) {
  const int e   = blockIdx.x;
  float* base   = wp + e * EXP_STRIDE;
  const int tid = threadIdx.x;
  const int nt  = blockDim.x;

  // W1: true shape [2][20] + bias row at k=2; padded to K=4 x 32 cols
  for (int i = tid; i < 4 * 32; i += nt) {
    int k = i >> 5, col = i & 31;
    float v = 0.0f;
    if (col < 20) {
      if (k < 2)       v = W1[e * 40 + k * 20 + col];
      else if (k == 2) v = b1[e * 20 + col];
    }
    base[W1_OFF + ((k >> 1) * 32 + col) * 2 + (k & 1)] = v;
  }
  // W2..W4: true shape [20][20] + bias row at k=20; padded to K=32 x 32 cols
  const float* Ws[3] = {W2, W3, W4};
  const float* Bs[3] = {b2, b3, b4};
  const int   Woff[3] = {W2_OFF, W3_OFF, W4_OFF};
  for (int l = 0; l < 3; ++l) {
    for (int i = tid; i < 32 * 32; i += nt) {
      int k = i >> 5, col = i & 31;
      float v = 0.0f;
      if (col < 20) {
        if (k < 20)       v = Ws[l][e * 400 + k * 20 + col];
        else if (k == 20) v = Bs[l][e * 20 + col];
      }
      base[Woff[l] + ((k >> 1) * 32 + col) * 2 + (k & 1)] = v;
    }
  }
  // W5: true shape [20][1] + bias at k=20; padded to K=32 x 16 cols (col 0)
  for (int i = tid; i < 32 * 16; i += nt) {
    int k = i >> 4, col = i & 15;
    float v = 0.0f;
    if (col == 0) {
      if (k < 20)       v = W5[e * 20 + k];
      else if (k == 20) v = b5[e];
    }
    base[W5_OFF + ((k >> 1) * 16 + col) * 2 + (k & 1)] = v;
  }
}

// ---------------------------------------------------------------------------
// Phase 1: per-tile evaluation with fp32 WMMA.
// ---------------------------------------------------------------------------
__global__ __launch_bounds__(32 * WPB) void moe_eval(
    const float* __restrict__ x, const float* __restrict__ wp,
    float* __restrict__ out, int ntiles) {
  __shared__ float hbuf[WPB][16][34];   // [wave][sample-row][hid], 34 for bank spread
  __shared__ int   idbuf[WPB][16];      // per-sample expert id

  const int wave = threadIdx.x >> 5;
  const int lane = threadIdx.x & 31;
  const int row  = lane & 15;           // sample-row / col index
  const int hi   = lane >> 4;           // 0: K-pair 0 / D rows 0-7, 1: K-pair 1 / rows 8-15
  const int tile = blockIdx.x * WPB + wave;
  if (tile >= ntiles) return;           // wave-uniform

  const int n = tile * 16 + row;
  const v2f xv = *(const v2f*)(x + 2 * n);
  const int ci = classify(xv.x);
  const int ri = classify(xv.y);
  const int my_e = (ci < 0 || ri < 0) ? -1 : ci * 4 + ri;  // e = ci*4 + ri
  if (hi == 0) idbuf[wave][row] = my_e;
  lds_order();

  // wave-uniform mask of experts present in this tile, + per-lane row ids
  unsigned present = 0;
  int rid[8];
#pragma unroll
  for (int r = 0; r < 16; ++r) {
    int id = idbuf[wave][r];
    if (id >= 0) present |= 1u << id;
  }
#pragma unroll
  for (int j = 0; j < 8; ++j) rid[j] = idbuf[wave][hi * 8 + j];

  // present is identical across lanes: force into SGPR -> scalar s_cbranch.
  const unsigned present_s =
      (unsigned)__builtin_amdgcn_readfirstlane((int)present);

  // Layer-1 A operand: 16x4 fp32, K = {x0, x1, 1.0(bias slot), 0}
  const v2f a1 = {hi ? 1.0f : xv.x, hi ? 0.0f : xv.y};

  // hi-column override: lanes with row==4 own activation col 20 (bias slot),
  // which must read 1.0 for the next layer's k=20 bias row.
  const bool bias_col = (row == 4);

  v8f acc = {};                         // masked output accumulator (col 0 lanes)

  float(*h)[34] = hbuf[wave];

  for (int e = 0; e < 16; ++e) {
    if (!((present_s >> e) & 1u)) continue;   // scalar-uniform skip
    const float* base = wp + e * EXP_STRIDE;

    // ---- layer 1: [16x4] x [4x32], bias via A k=2 slot ----
    v8f clo = {}, chi = {};
    {
      const float* w = base + W1_OFF + hi * 64;  // k2 = hi
      v2f blo = *(const v2f*)(w + row * 2);
      v2f bhi = *(const v2f*)(w + (16 + row) * 2);
      clo = wmma_f32(a1, blo, clo);
      chi = wmma_f32(a1, bhi, chi);
    }
    lds_order();  // WAR order vs previous expert's layer-5 LDS reads
#pragma unroll
    for (int j = 0; j < 8; ++j) {
      h[hi * 8 + j][row]      = silu(clo[j]);
      h[hi * 8 + j][16 + row] = bias_col ? 1.0f : silu(chi[j]);
    }
    lds_order();

    // ---- layers 2..4: [16x24] x [24x32], bias via k=20 row ----
    const int Woff[3] = {W2_OFF, W3_OFF, W4_OFF};
#pragma unroll
    for (int l = 0; l < 3; ++l) {
      const float* w = base + Woff[l];
      v8f lo = {}, hic = {};
#pragma unroll
      for (int ks = 0; ks < NKS; ++ks) {
        v2f a = *(const v2f*)(&h[row][4 * ks + 2 * hi]);   // 8B-aligned ds_load_b64
        const float* wk = w + ((2 * ks + hi) * 32) * 2;
        v2f blo = *(const v2f*)(wk + row * 2);
        v2f bhi = *(const v2f*)(wk + (16 + row) * 2);
        lo  = wmma_f32(a, blo, lo);
        hic = wmma_f32(a, bhi, hic);
      }
      lds_order();  // WAR order: this layer's A reads before overwrite
#pragma unroll
      for (int j = 0; j < 8; ++j) {
        h[hi * 8 + j][row]      = silu(lo[j]);
        h[hi * 8 + j][16 + row] = bias_col ? 1.0f : silu(hic[j]);
      }
      lds_order();
    }

    // ---- layer 5: [16x24] x [24x16], bias via k=20 row; col 0 meaningful ----
    v8f c5 = {};
#pragma unroll
    for (int ks = 0; ks < NKS; ++ks) {
      v2f a = *(const v2f*)(&h[row][4 * ks + 2 * hi]);
      const float* wk = base + W5_OFF + ((2 * ks + hi) * 16) * 2;
      v2f b = *(const v2f*)(wk + row * 2);
      c5 = wmma_f32(a, b, c5);
    }
    // D layout: lane holds col=row, rows hi*8+j. Col 0 lives in lanes 0 / 16.
#pragma unroll
    for (int j = 0; j < 8; ++j)
      if (rid[j] == e) acc[j] += c5[j];
  }

  if (row == 0) {
#pragma unroll
    for (int j = 0; j < 8; ++j) out[tile * 16 + hi * 8 + j] = acc[j];
  }
}

// ---------------------------------------------------------------------------
extern "C" void kernel_launch(void* const* d_in, const int* in_sizes, int n_in,
                              void* d_out, int out_size, void* d_ws, size_t ws_size,
                              hipStream_t stream) {
  const float* x  = (const float*)d_in[0];
  const float* W1 = (const float*)d_in[1];
  const float* b1 = (const float*)d_in[2];
  const float* W2 = (const float*)d_in[3];
  const float* b2 = (const float*)d_in[4];
  const float* W3 = (const float*)d_in[5];
  const float* b3 = (const float*)d_in[6];
  const float* W4 = (const float*)d_in[7];
  const float* b4 = (const float*)d_in[8];
  const float* W5 = (const float*)d_in[9];
  const float* b5 = (const float*)d_in[10];

  float* wp  = (float*)d_ws;     // needs 16 * 4096 * 4 = 256 KB
  float* out = (float*)d_out;

  const int N      = in_sizes[0] / 2;
  const int ntiles = N / 16;
  const int blocks = (ntiles + WPB - 1) / WPB;

  prep_weights<<<16, 256, 0, stream>>>(W1, b1, W2, b2, W3, b3, W4, b4, W5, b5, wp);
  moe_eval<<<blocks, 32 * WPB, 0, stream>>>(x, wp, out, ntiles);
}